// Group_77412490543267
// MI455X (gfx1250) — compile-verified
//
#include <hip/hip_runtime.h>
#include <hip/hip_bf16.h>

#define N_POINTS   65536
#define NUM_GROUP  2048
#define GROUP_SIZE 32
#define CTILES     (NUM_GROUP / 16)   // 128 blocks, 16 centers each

typedef __attribute__((ext_vector_type(2))) float v2f;
typedef __attribute__((ext_vector_type(8))) float v8f;

__device__ __forceinline__ void argmax_combine(float& v, int& i, float ov, int oi) {
  // jnp.argmax semantics: larger value wins; on tie, lower index wins
  if (ov > v || (ov == v && oi < i)) { v = ov; i = oi; }
}

// ---------------------------------------------------------------------------
// Kernel 1: farthest point sampling.  One workgroup, min_d[] resident in the
// 320KB CDNA5 LDS; 2048 serial iterations of (distance, min, block argmax).
// ---------------------------------------------------------------------------
__global__ __launch_bounds__(1024)
void fps_kernel(const float4* __restrict__ exyz, int* __restrict__ cidx) {
  extern __shared__ float min_d[];            // N_POINTS floats = 256KB
  __shared__ float red_v[32];
  __shared__ int   red_i[32];
  __shared__ int   s_cur;
  __shared__ float s_cx, s_cy, s_cz;

  const int tid  = threadIdx.x;
  const int lane = tid & 31;
  const int wid  = tid >> 5;

  for (int p = tid; p < N_POINTS; p += 1024) min_d[p] = 3.4e38f;
  if (tid == 0) s_cur = 0;
  __syncthreads();

  for (int it = 0; it < NUM_GROUP; ++it) {
    if (tid == 0) {
      cidx[it] = s_cur;
      float4 c = exyz[s_cur];
      s_cx = c.y; s_cy = c.z; s_cz = c.w;   // xyz = exyz[..., 1:]
    }
    __syncthreads();
    const float cx = s_cx, cy = s_cy, cz = s_cz;

    float best = -1.0f;
    int   bi   = 0x7fffffff;
    #pragma unroll 4
    for (int j = 0; j < N_POINTS / 1024; ++j) {
      const int p = tid + j * 1024;            // coalesced float4 loads
      float4 q = exyz[p];
      float dx = q.y - cx, dy = q.z - cy, dz = q.w - cz;
      float d  = dx * dx + dy * dy + dz * dz;
      float m  = fminf(min_d[p], d);
      min_d[p] = m;
      if (m > best) { best = m; bi = p; }      // ascending p -> stable tie
    }
    // wave32 argmax
    #pragma unroll
    for (int off = 16; off > 0; off >>= 1) {
      float ov = __shfl_down(best, off, 32);
      int   oi = __shfl_down(bi,   off, 32);
      argmax_combine(best, bi, ov, oi);
    }
    if (lane == 0) { red_v[wid] = best; red_i[wid] = bi; }
    __syncthreads();
    if (wid == 0) {
      best = red_v[lane]; bi = red_i[lane];
      #pragma unroll
      for (int off = 16; off > 0; off >>= 1) {
        float ov = __shfl_down(best, off, 32);
        int   oi = __shfl_down(bi,   off, 32);
        argmax_combine(best, bi, ov, oi);
      }
      if (lane == 0) s_cur = bi;
    }
    __syncthreads();
  }
}

// ---------------------------------------------------------------------------
// Kernel 2: per-center 32-NN via V_WMMA_F32_16X16X4_F32 score tiles
//   score = [-2cx,-2cy,-2cz,1] . [x,y,z,|x|^2] = d2 - |c|^2  (row-constant
//   offset dropped: does not change per-center top-k ordering).
// Block = 16 centers, 256 threads (8 waves). 256-point chunks through LDS,
// register-resident sorted top-32 per (center, stripe) thread, wave-wide
// cursor-merge at the end.  Fragments are built branchlessly so the hot loop
// has no EXEC-divergent regions: one global_load_b128 + cndmask selects.
// ---------------------------------------------------------------------------
__global__ __launch_bounds__(256)
void group_kernel(const float4* __restrict__ exyz, const int* __restrict__ cidx,
                  float* __restrict__ out) {
  extern __shared__ char smem_raw[];
  float* s_scores = (float*)smem_raw;             // 16 x 256
  float* cand_v   = s_scores + 16 * 256;          // 16 x 512
  int*   cand_i   = (int*)(cand_v + 16 * 512);    // 16 x 512
  int*   s_outidx = cand_i + 16 * 512;            // 16 x 32

  const int  tid  = threadIdx.x;
  const int  lane = tid & 31;
  const int  wid  = tid >> 5;
  const bool hi   = (lane & 16) != 0;  // low half: K0,K1 | high half: K2,K3
  const int  l15  = lane & 15;
  const int  c0   = blockIdx.x * 16;

  // A fragment: 16x4 f32 (ISA layout: lanes0-15 hold K0,K1; lanes16-31 K2,K3)
  const int    ci = cidx[c0 + l15];
  const float4 cc = exyz[ci];
  v2f a;
  a.x = hi ? (-2.0f * cc.w) : (-2.0f * cc.y);
  a.y = hi ? 1.0f            : (-2.0f * cc.z);

  // register-resident sorted top-32 (ascending)
  float bv[GROUP_SIZE];
  int   bp[GROUP_SIZE];
  #pragma unroll
  for (int q = 0; q < GROUP_SIZE; ++q) { bv[q] = 3.4e38f; bp[q] = 0x7fffffff; }

  const int selc = tid >> 4;   // center 0..15
  const int sels = tid & 15;   // stripe 0..15

  for (int chunk = 0; chunk < N_POINTS / 256; ++chunk) {
    const int base = chunk * 256;
    // --- compute 16x256 score tile: 2 WMMAs per wave ---
    #pragma unroll
    for (int t = 0; t < 2; ++t) {
      const int tb = wid * 32 + t * 16;
      const float4 e  = exyz[base + tb + l15];        // one b128 load, no branch
      const float  n2 = e.y * e.y + e.z * e.z + e.w * e.w;
      v2f b;
      b.x = hi ? e.w : e.y;                           // K2=z   | K0=x
      b.y = hi ? n2  : e.z;                           // K3=|x|²| K1=y
      v8f acc = {};
      acc = __builtin_amdgcn_wmma_f32_16x16x4_f32(
          /*neg_a=*/false, a, /*neg_b=*/false, b,
          /*c_mod=*/(short)0, acc, /*reuse_a=*/false, /*reuse_b=*/false);
      const int rbase = hi ? 8 : 0;    // D layout: lanes16-31 hold M=v+8
      #pragma unroll
      for (int v = 0; v < 8; ++v)
        s_scores[(rbase + v) * 256 + tb + l15] = acc[v];
    }
    __syncthreads();
    // --- selection: thread (selc,sels) scans stride-16 stripe ---
    #pragma unroll
    for (int j = 0; j < 16; ++j) {
      const int colp = j * 16 + sels;
      float d = s_scores[selc * 256 + colp];
      int   p = base + colp;
      if (d < bv[GROUP_SIZE - 1]) {        // strict < : stable, lowest idx kept
        #pragma unroll
        for (int q = 0; q < GROUP_SIZE; ++q) {
          const bool sw = d < bv[q];
          const float tv = bv[q]; const int tp = bp[q];
          bv[q] = sw ? d : tv;  bp[q] = sw ? p : tp;
          d = sw ? tv : d;      p = sw ? tp : p;
        }
      }
    }
    __syncthreads();
  }

  // --- dump sorted per-thread lists ---
  #pragma unroll
  for (int q = 0; q < GROUP_SIZE; ++q) {
    cand_v[selc * 512 + sels * 32 + q] = bv[q];
    cand_i[selc * 512 + sels * 32 + q] = bp[q];
  }
  __syncthreads();

  // --- merge 32 sorted runs of 16 per center (wave w: centers 2w, 2w+1) ---
  for (int cc2 = 0; cc2 < 2; ++cc2) {
    const int c = wid * 2 + cc2;
    const int mybase = c * 512 + lane * 16;   // each lane owns one sorted run
    int   cur = 0;
    float hv = cand_v[mybase];
    int   hp = cand_i[mybase];
    for (int k = 0; k < GROUP_SIZE; ++k) {
      float v = hv; int pi = hp; int ln = lane;
      #pragma unroll
      for (int off = 16; off > 0; off >>= 1) {
        const float ov  = __shfl_down(v,  off, 32);
        const int   opi = __shfl_down(pi, off, 32);
        const int   oln = __shfl_down(ln, off, 32);
        if (ov < v || (ov == v && opi < pi)) { v = ov; pi = opi; ln = oln; }
      }
      if (lane == 0) s_outidx[c * 32 + k] = pi;
      const int wln = __shfl(ln, 0, 32);
      if (lane == wln) {
        ++cur;
        if (cur < 16) { hv = cand_v[mybase + cur]; hp = cand_i[mybase + cur]; }
        else          { hv = 3.4e38f; hp = 0x7fffffff; }
      }
    }
  }
  __syncthreads();

  // --- gather outputs: neighborhood (2048,32,4) then centers (2048,4) ---
  float4* out4 = (float4*)out;
  for (int e = tid; e < 16 * GROUP_SIZE; e += 256) {
    const int c = e >> 5, k = e & 31;
    out4[(size_t)(c0 + c) * GROUP_SIZE + k] = exyz[s_outidx[c * 32 + k]];
  }
  if (tid < 16) {
    const int g = c0 + tid;
    out4[(size_t)NUM_GROUP * GROUP_SIZE + g] = exyz[cidx[g]];
  }
}

extern "C" void kernel_launch(void* const* d_in, const int* in_sizes, int n_in,
                              void* d_out, int out_size, void* d_ws, size_t ws_size,
                              hipStream_t stream) {
  const float4* exyz = (const float4*)d_in[0];
  float* out  = (float*)d_out;
  int*   cidx = (int*)d_ws;                     // 2048 int32 in scratch

  // FPS: one workgroup, 256KB dynamic LDS for min_d
  hipLaunchKernelGGL(fps_kernel, dim3(1), dim3(1024),
                     N_POINTS * sizeof(float), stream, exyz, cidx);

  // Grouping: 128 blocks x 256 threads
  const size_t lds = (16 * 256 + 16 * 512) * sizeof(float)  // scores + cand_v
                   + (16 * 512 + 16 * 32) * sizeof(int);    // cand_i + outidx
  hipLaunchKernelGGL(group_kernel, dim3(CTILES), dim3(256),
                     lds, stream, exyz, cidx, out);
}